// WNRMSELoss_17085379903763
// MI455X (gfx1250) — compile-verified
//
#include <hip/hip_runtime.h>
#include <hip/hip_bf16.h>
#include <math.h>

typedef __attribute__((ext_vector_type(2))) float v2f;
typedef __attribute__((ext_vector_type(4))) float v4f;
typedef __attribute__((ext_vector_type(8))) float v8f;

// ds_swizzle_b32 bitmask mode: out[lane] = src[lane ^ K]  (offset = and:0x1f | xor:K<<10)
#define SWZ_XOR_F32(x, K)                                        \
  __int_as_float(__builtin_amdgcn_ds_swizzle(                    \
      __float_as_int(x), (0x1f | ((K) << 10))))

// Full 32-lane (wave32) sum using one V_WMMA_F32_16X16X4_F32.
// A layout (16x4 f32): lanes 0-15 supply A[m][0..1], lanes 16-31 supply A[m][2..3].
// With a[0]=a[1]=p and B = 0.5 everywhere (constant B => B-layout-free):
//   D[m][n] = 0.5*(2*p_m + 2*p_{m+16}) = p_m + p_{m+16}
// D layout: VGPR r holds rows r (lanes 0-15) and r+8 (lanes 16-31), so
// sum(d[0..7]) yields the low-half row sums on lanes 0-15 and the high-half
// on lanes 16-31; one lane-xor-16 swizzle completes the full-wave sum.
static __device__ __forceinline__ float wave_sum_wmma(float p) {
  v2f a; a[0] = p;    a[1] = p;
  v2f b; b[0] = 0.5f; b[1] = 0.5f;
  v8f c = {0.f, 0.f, 0.f, 0.f, 0.f, 0.f, 0.f, 0.f};
  v8f d = __builtin_amdgcn_wmma_f32_16x16x4_f32(
      /*neg_a=*/false, a, /*neg_b=*/false, b,
      /*c_mod=*/(short)0, c, /*reuse_a=*/false, /*reuse_b=*/false);
  float s = d[0] + d[1] + d[2] + d[3] + d[4] + d[5] + d[6] + d[7];
  s += SWZ_XOR_F32(s, 16);
  return s;
}

static __device__ __forceinline__ float wave_max(float x) {
  x = fmaxf(x, SWZ_XOR_F32(x, 1));
  x = fmaxf(x, SWZ_XOR_F32(x, 2));
  x = fmaxf(x, SWZ_XOR_F32(x, 4));
  x = fmaxf(x, SWZ_XOR_F32(x, 8));
  x = fmaxf(x, SWZ_XOR_F32(x, 16));
  return x;
}

static __device__ __forceinline__ float wave_min(float x) {
  x = fminf(x, SWZ_XOR_F32(x, 1));
  x = fminf(x, SWZ_XOR_F32(x, 2));
  x = fminf(x, SWZ_XOR_F32(x, 4));
  x = fminf(x, SWZ_XOR_F32(x, 8));
  x = fminf(x, SWZ_XOR_F32(x, 16));
  return x;
}

// Pass 1: one 256-thread block per sample row (N = 2048).
// All 12 non-temporal b128 loads are issued with independent accumulators
// (max MLP), then a single combined reduction phase with one barrier.
__global__ __launch_bounds__(256) void wnrmse_pass1(
    const float* __restrict__ o1, const float* __restrict__ t1,
    const float* __restrict__ o2, const float* __restrict__ t2,
    const float* __restrict__ o3, const float* __restrict__ t3,
    float* __restrict__ per_sample) {
  constexpr int N = 2048;
  const int tid  = threadIdx.x;
  const int lane = tid & 31;
  const int wid  = tid >> 5;
  const size_t base = (size_t)blockIdx.x * (size_t)N;

  const float* const os[3] = {o1 + base, o2 + base, o3 + base};
  const float* const ts[3] = {t1 + base, t2 + base, t3 + base};
  const float        wt[3] = {0.5f, 0.25f, 0.25f};

  float ssq[3], tmx[3], tmn[3];
#pragma unroll
  for (int p = 0; p < 3; ++p) {
    ssq[p] = 0.0f;
    tmx[p] = -__builtin_inff();
    tmn[p] =  __builtin_inff();
  }

  // Streaming phase: 2 iterations x 3 pairs x (o,t) = 12 b128 NT loads/thread.
#pragma unroll
  for (int i = 0; i < 2; ++i) {
    const int off = i * 1024 + tid * 4;  // 256 threads x float4 = 1024 floats/iter
#pragma unroll
    for (int p = 0; p < 3; ++p) {
      v4f ov = __builtin_nontemporal_load((const v4f*)(os[p] + off));
      v4f tv = __builtin_nontemporal_load((const v4f*)(ts[p] + off));
      v4f dv = ov - tv;
      ssq[p] += dv[0] * dv[0] + dv[1] * dv[1] + dv[2] * dv[2] + dv[3] * dv[3];
      tmx[p] = fmaxf(tmx[p], fmaxf(fmaxf(tv[0], tv[1]), fmaxf(tv[2], tv[3])));
      tmn[p] = fminf(tmn[p], fminf(fminf(tv[0], tv[1]), fminf(tv[2], tv[3])));
    }
  }

  // Single reduction phase (all 32 lanes active -> EXEC all ones for WMMA).
#pragma unroll
  for (int p = 0; p < 3; ++p) ssq[p] = wave_sum_wmma(ssq[p]);
#pragma unroll
  for (int p = 0; p < 3; ++p) {
    tmx[p] = wave_max(tmx[p]);
    tmn[p] = wave_min(tmn[p]);
  }

  __shared__ float s_ssq[3][8];
  __shared__ float s_max[3][8];
  __shared__ float s_min[3][8];
  if (lane == 0) {
#pragma unroll
    for (int p = 0; p < 3; ++p) {
      s_ssq[p][wid] = ssq[p];
      s_max[p][wid] = tmx[p];
      s_min[p][wid] = tmn[p];
    }
  }
  __syncthreads();

  if (tid == 0) {
    float acc = 0.0f;
#pragma unroll
    for (int p = 0; p < 3; ++p) {
      float S = 0.0f, MX = -__builtin_inff(), MN = __builtin_inff();
#pragma unroll
      for (int w = 0; w < 8; ++w) {
        S += s_ssq[p][w];
        MX = fmaxf(MX, s_max[p][w]);
        MN = fminf(MN, s_min[p][w]);
      }
      const float rmse = sqrtf(S * (1.0f / (float)N));
      acc += wt[p] * rmse / (MX - MN);
    }
    per_sample[blockIdx.x] = acc;
  }
}

// Pass 2: single block reduces the 4096 per-sample losses deterministically.
__global__ __launch_bounds__(256) void wnrmse_pass2(
    const float* __restrict__ per_sample, float* __restrict__ out, int n) {
  const int tid  = threadIdx.x;
  const int lane = tid & 31;
  const int wid  = tid >> 5;

  float s = 0.0f;
  for (int i = tid; i < n; i += 256) s += per_sample[i];  // n % 256 == 0: all lanes active
  s = wave_sum_wmma(s);

  __shared__ float sw[8];
  if (lane == 0) sw[wid] = s;
  __syncthreads();
  if (tid == 0) {
    float S = 0.0f;
#pragma unroll
    for (int w = 0; w < 8; ++w) S += sw[w];
    out[0] = S / (float)n;
  }
}

extern "C" void kernel_launch(void* const* d_in, const int* in_sizes, int n_in,
                              void* d_out, int out_size, void* d_ws, size_t ws_size,
                              hipStream_t stream) {
  const float* o1 = (const float*)d_in[0];
  const float* t1 = (const float*)d_in[1];
  const float* o2 = (const float*)d_in[2];
  const float* t2 = (const float*)d_in[3];
  const float* o3 = (const float*)d_in[4];
  const float* t3 = (const float*)d_in[5];

  const int N = 2048;
  const int B = in_sizes[0] / N;  // 4096

  float* per_sample = (float*)d_ws;  // B floats of scratch

  wnrmse_pass1<<<B, 256, 0, stream>>>(o1, t1, o2, t2, o3, t3, per_sample);
  wnrmse_pass2<<<1, 256, 0, stream>>>(per_sample, (float*)d_out, B);
}